// LocalConvModule_86337432584585
// MI455X (gfx1250) — compile-verified
//
#include <hip/hip_runtime.h>
#include <hip/hip_bf16.h>
#include <math.h>

typedef __attribute__((ext_vector_type(2))) float v2f;
typedef __attribute__((ext_vector_type(8))) float v8f;

#define N_IMG 64
#define C_CH 512
#define HW 1024
#define TOPK 16
#define CONCAT_W (TOPK * C_CH + HW)              // 9216
#define OFF_CONCAT 0
#define OFF_MASK ((size_t)N_IMG * CONCAT_W)      // 589824 floats
#define OFF_OUT (OFF_MASK + (size_t)N_IMG * HW)  // 655360 floats

// ---------------------------------------------------------------------------
// Kernel 1: t[n,hw] = sum_c x[n,c,hw] * w[c] via V_WMMA_F32_16X16X4_F32.
// One wave handles 16 hw positions of one image; K-loop over C in steps of 4.
// A (16x4, all rows = w[c0..c0+3]):  VGPR0 = {K0 | K2}, VGPR1 = {K1 | K3}
// B (4x16):                          VGPR0 = {K0 row | K2 row}, VGPR1 = {K1 | K3}
// Every row of D equals t; lanes 0..15 read acc[0] = D[M=0, N=lane].
// ---------------------------------------------------------------------------
__global__ __launch_bounds__(256) void dot_wmma_kernel(
    const float* __restrict__ x, const float* __restrict__ w,
    float* __restrict__ t) {
  const int lane = threadIdx.x & 31;
  const int wave = (blockIdx.x * (blockDim.x >> 5)) + (threadIdx.x >> 5); // 0..4095
  const int n = wave >> 6;        // /64 tiles per image
  const int hw0 = (wave & 63) << 4;
  const int hi = (lane >= 16) ? 1 : 0;  // upper lane half holds K=2,3
  const int col = lane & 15;
  const float* __restrict__ xn = x + ((size_t)n * C_CH * HW);

  v8f acc = {};
  for (int c0 = 0; c0 < C_CH; c0 += 4) {
    const int k0 = c0 + (hi ? 2 : 0);
    const int k1 = k0 + 1;
    v2f a, b;
    a.x = w[k0];
    a.y = w[k1];
    b.x = xn[(size_t)k0 * HW + hw0 + col];
    b.y = xn[(size_t)k1 * HW + hw0 + col];
    acc = __builtin_amdgcn_wmma_f32_16x16x4_f32(
        /*neg_a=*/false, a, /*neg_b=*/false, b,
        /*c_mod=*/(short)0, acc, /*reuse_a=*/false, /*reuse_b=*/false);
  }
  if (lane < 16) t[n * HW + hw0 + col] = acc[0];
}

// ---------------------------------------------------------------------------
// Kernel 2: per-image softmax over HW, iterative top-16 argmax, rank-sort of
// the selected indices, and straight-through mask st = (hard - p) + p written
// to both the concat tail and the st_mask_4d output region.
// One 256-thread block per image.
// ---------------------------------------------------------------------------
__global__ __launch_bounds__(256) void softmax_topk_kernel(
    const float* __restrict__ t, float* __restrict__ out,
    int* __restrict__ idx_out) {
  __shared__ float tv[HW];
  __shared__ float sv[256];
  __shared__ int si[256];
  __shared__ unsigned char mk[HW];
  __shared__ int list[TOPK];
  __shared__ int sorted[TOPK];
  __shared__ float s_max, s_sum;

  const int n = blockIdx.x;
  const int tid = threadIdx.x;
  const float* __restrict__ tn = t + (size_t)n * HW;

  for (int j = 0; j < 4; ++j) {
    const int i = tid + j * 256;
    tv[i] = tn[i];
    mk[i] = 0;
  }
  __syncthreads();

  // max reduction
  float m = -INFINITY;
  for (int j = 0; j < 4; ++j) m = fmaxf(m, tv[tid + j * 256]);
  sv[tid] = m;
  __syncthreads();
  for (int s = 128; s > 0; s >>= 1) {
    if (tid < s) sv[tid] = fmaxf(sv[tid], sv[tid + s]);
    __syncthreads();
  }
  if (tid == 0) s_max = sv[0];
  __syncthreads();
  const float mx = s_max;

  // exp and sum
  float lsum = 0.0f;
  for (int j = 0; j < 4; ++j) {
    const int i = tid + j * 256;
    const float e = expf(tv[i] - mx);
    tv[i] = e;
    lsum += e;
  }
  sv[tid] = lsum;
  __syncthreads();
  for (int s = 128; s > 0; s >>= 1) {
    if (tid < s) sv[tid] += sv[tid + s];
    __syncthreads();
  }
  if (tid == 0) s_sum = sv[0];
  __syncthreads();
  const float sum = s_sum;

  // iterative top-16 argmax (exp is monotone in t; ties -> smaller index)
  for (int it = 0; it < TOPK; ++it) {
    float bv = -INFINITY;
    int bi = 1 << 30;
    for (int j = 0; j < 4; ++j) {
      const int i = tid + j * 256;
      if (!mk[i]) {
        const float v = tv[i];
        if (v > bv || (v == bv && i < bi)) { bv = v; bi = i; }
      }
    }
    sv[tid] = bv;
    si[tid] = bi;
    __syncthreads();
    for (int s = 128; s > 0; s >>= 1) {
      if (tid < s) {
        const float ov = sv[tid + s];
        const int oi = si[tid + s];
        if (ov > sv[tid] || (ov == sv[tid] && oi < si[tid])) {
          sv[tid] = ov;
          si[tid] = oi;
        }
      }
      __syncthreads();
    }
    if (tid == 0) {
      list[it] = si[0];
      mk[si[0]] = 1;
    }
    __syncthreads();
  }

  // rank-sort the 16 (distinct) indices ascending
  if (tid < TOPK) {
    const int v = list[tid];
    int r = 0;
    for (int j = 0; j < TOPK; ++j) r += (list[j] < v);
    sorted[r] = v;
  }
  __syncthreads();
  if (tid < TOPK) idx_out[n * TOPK + tid] = sorted[tid];

  // st mask writes: concat tail [n, 8192..9215] and st_mask_4d region
  float* __restrict__ concat_mask = out + (size_t)n * CONCAT_W + TOPK * C_CH;
  float* __restrict__ mask4d = out + OFF_MASK + (size_t)n * HW;
  for (int j = 0; j < 4; ++j) {
    const int i = tid + j * 256;
    const float p = tv[i] / sum;
    const float hard = mk[i] ? 1.0f : 0.0f;
    const float st = (hard - p) + p;  // matches reference arithmetic exactly
    concat_mask[i] = st;
    mask4d[i] = st;
  }
}

// ---------------------------------------------------------------------------
// Kernel 3: out[n,c,hw] = st[n,hw] * x[n,c,hw], float4-vectorized stream.
// ---------------------------------------------------------------------------
__global__ __launch_bounds__(256) void scale_kernel(
    const float4* __restrict__ x4, const float4* __restrict__ st4,
    float4* __restrict__ out4) {
  const size_t i = (size_t)blockIdx.x * blockDim.x + threadIdx.x; // 8388608 f4
  const int n = (int)(i >> 17);  // / (512*256)
  const int hw4 = (int)(i & 255);
  const float4 s = st4[n * 256 + hw4];
  const float4 v = x4[i];
  float4 o;
  o.x = v.x * s.x;
  o.y = v.y * s.y;
  o.z = v.z * s.z;
  o.w = v.w * s.w;
  out4[i] = o;
}

// ---------------------------------------------------------------------------
// Kernel 4: out_efficient[n, k*C + c] = out[n, c, idx_sorted[n,k]]
// Reads from the just-written `out` region (hot in the 192MB L2).
// ---------------------------------------------------------------------------
__global__ __launch_bounds__(256) void gather_kernel(
    const float* __restrict__ outC, const int* __restrict__ idxs,
    float* __restrict__ concat) {
  const int i = blockIdx.x * 256 + threadIdx.x;  // 64*16*512 = 524288
  const int c = i & (C_CH - 1);
  const int k = (i >> 9) & (TOPK - 1);
  const int n = i >> 13;
  const int hw = idxs[n * TOPK + k];
  const float v = outC[(((size_t)n * C_CH + c) << 10) + hw];
  concat[(size_t)n * CONCAT_W + k * C_CH + c] = v;
}

extern "C" void kernel_launch(void* const* d_in, const int* in_sizes, int n_in,
                              void* d_out, int out_size, void* d_ws,
                              size_t ws_size, hipStream_t stream) {
  (void)in_sizes; (void)n_in; (void)out_size; (void)ws_size;
  const float* x = (const float*)d_in[0];
  const float* w = (const float*)d_in[1];
  float* out = (float*)d_out;

  float* t = (float*)d_ws;                             // 64*1024 floats
  int* idx = (int*)((char*)d_ws + N_IMG * HW * sizeof(float));

  // 1) channel dot product via WMMA: 4096 waves, 8 waves/block
  dot_wmma_kernel<<<512, 256, 0, stream>>>(x, w, t);

  // 2) softmax + top-16 + st_mask per image
  softmax_topk_kernel<<<N_IMG, 256, 0, stream>>>(t, out, idx);

  // 3) masked multiply stream (out region)
  scale_kernel<<<32768, 256, 0, stream>>>(
      (const float4*)x, (const float4*)(out + OFF_MASK),
      (float4*)(out + OFF_OUT));

  // 4) gather selected columns into concat head
  gather_kernel<<<2048, 256, 0, stream>>>(out + OFF_OUT, idx, out);
}